// PhysicsLoss_12043088298445
// MI455X (gfx1250) — compile-verified
//
#include <hip/hip_runtime.h>
#include <math.h>

typedef float v2f __attribute__((ext_vector_type(2)));
typedef float v4f __attribute__((ext_vector_type(4)));
typedef float v8f __attribute__((ext_vector_type(8)));

#define LROW 1024
#define K1_BLOCKS 2048
#define K1_THREADS 256
#define WAVES_PER_BLOCK 8

// ---- ordered-int encoding for exact float max atomics (deterministic) ----
__device__ __forceinline__ unsigned flip_f32(float f) {
  unsigned u = __float_as_uint(f);
  return (u & 0x80000000u) ? ~u : (u | 0x80000000u);
}
__device__ __forceinline__ float unflip_f32(unsigned k) {
  unsigned b = (k & 0x80000000u) ? (k & 0x7FFFFFFFu) : ~k;
  return __uint_as_float(b);
}

// ---- full-wave (32 lane) f32 sum via two V_WMMA_F32_16X16X4_F32 ----
// A layout (ISA 7.12.2): lanes 0-15 hold A[m][0]/A[m][1] in VGPR0/1,
// lanes 16-31 hold A[m][2]/A[m][3]. B = all-ones is layout-invariant.
// WMMA #1: D[m][n] = x[m] + x[m+16]; sum D's 8 VGPRs per lane;
// WMMA #2 broadcasts the total to every lane.
__device__ __forceinline__ float wave_sum_f32(float x) {
  v2f a; a[0] = x;    a[1] = 0.0f;
  v2f b; b[0] = 1.0f; b[1] = 1.0f;
  v8f c = {0.f, 0.f, 0.f, 0.f, 0.f, 0.f, 0.f, 0.f};
  v8f d = __builtin_amdgcn_wmma_f32_16x16x4_f32(false, a, false, b,
                                                (short)0, c, false, false);
  float s = ((d[0] + d[1]) + (d[2] + d[3])) + ((d[4] + d[5]) + (d[6] + d[7]));
  v2f a2; a2[0] = s; a2[1] = 0.0f;
  v8f d2 = __builtin_amdgcn_wmma_f32_16x16x4_f32(false, a2, false, b,
                                                 (short)0, c, false, false);
  return d2[0];
}

__global__ void k0_init(unsigned* gmax_key) {
  if (threadIdx.x == 0 && blockIdx.x == 0) *gmax_key = flip_f32(-INFINITY);
}

// ---- Kernel 1: one pass over waveform. One wave per row. ----
__global__ __launch_bounds__(K1_THREADS) void k1_rows(
    const float* __restrict__ wf, const float* __restrict__ depth,
    float* __restrict__ e_raw, float* __restrict__ d2_partial,
    unsigned* __restrict__ gmax_key, int B) {
  const int lane  = threadIdx.x & 31;
  const int wave  = threadIdx.x >> 5;
  const int wid0  = blockIdx.x * WAVES_PER_BLOCK + wave;
  const int wstep = gridDim.x * WAVES_PER_BLOCK;
  const float inv_sigma = 1.0f / 5.0f;  // sigma = WIN/4 = 5
  const int rotL = (lane + 31) & 31;    // source lane for left neighbor
  const int rotR = (lane + 1) & 31;     // source lane for right neighbor

  float d2acc = 0.0f;
  float mx = -INFINITY;

  for (int row = wid0; row < B; row += wstep) {
    const v4f* r4 = (const v4f*)(wf + (size_t)row * LROW);
    // tp is identical in all lanes of this wave -> force to SGPR so the
    // per-band window test below becomes a scalar branch (no EXEC churn).
    const float tp = __uint_as_float(
        __builtin_amdgcn_readfirstlane(__float_as_uint(depth[row] * 10.0f)));
    v4f v[8];
#pragma unroll
    for (int j = 0; j < 8; ++j)
      v[j] = __builtin_nontemporal_load(&r4[j * 32 + lane]);  // streamed once

    float en = 0.0f, wsm = 0.0f;
#pragma unroll
    for (int j = 0; j < 8; ++j) {
      const v4f q = v[j];
      const int i0 = (j * 32 + lane) * 4;

      // running max (always)
      mx = fmaxf(mx, fmaxf(fmaxf(q.x, q.y), fmaxf(q.z, q.w)));

      // Gaussian window + energy: this j-band covers samples
      // [j*128, j*128+127]; beyond ~±70 samples of tp the f32 Gaussian is
      // < 1e-38 and cannot perturb accumulators ~O(10). Scalar-skip far bands.
      const float lo = (float)(j * 128);
      if (tp > lo - 80.0f && tp < lo + 207.0f) {
        const float t0 = (float)i0;
        const float d0 = (t0 - tp) * inv_sigma;
        const float d1 = (t0 + 1.0f - tp) * inv_sigma;
        const float d2_ = (t0 + 2.0f - tp) * inv_sigma;
        const float d3 = (t0 + 3.0f - tp) * inv_sigma;
        const float g0 = __expf(-0.5f * d0 * d0);
        const float g1 = __expf(-0.5f * d1 * d1);
        const float g2 = __expf(-0.5f * d2_ * d2_);
        const float g3 = __expf(-0.5f * d3 * d3);
        wsm += (g0 + g1) + (g2 + g3);
        en  += ((q.x * q.x) * g0 + (q.y * q.y) * g1) +
               ((q.z * q.z) * g2 + (q.w * q.w) * g3);
      }

      // neighbor exchange: ONE rotate-shuffle per direction per j.
      // Lane 31 supplies the previous chunk's .w to lane 0; lane 0 supplies
      // the next chunk's .x to lane 31; all other lanes supply q.w / q.x.
      const float prevw = (j > 0) ? v[j - 1].w : 0.0f;
      const float nextx = (j < 7) ? v[j + 1].x : 0.0f;
      const float supL = (lane == 31) ? prevw : q.w;
      const float supR = (lane == 0) ? nextx : q.x;
      const float left  = __shfl(supL, rotL, 32);
      const float right = __shfl(supR, rotR, 32);

      const float e0 = (i0 == 0) ? (q.y - q.x)
                                 : (q.y - 2.0f * q.x + left);
      const float e1 = q.z - 2.0f * q.y + q.x;
      const float e2 = q.w - 2.0f * q.z + q.y;
      const float e3 = (i0 + 3 == LROW - 1) ? (q.z - q.w)
                                            : (right - 2.0f * q.w + q.z);
      d2acc += (e0 * e0 + e1 * e1) + (e2 * e2 + e3 * e3);
    }

    const float en_t = wave_sum_f32(en);   // EXEC is all-ones here
    const float ws_t = wave_sum_f32(wsm);
    if (lane == 0) e_raw[row] = en_t / ws_t;
  }

  const float d2_t = wave_sum_f32(d2acc);
#pragma unroll
  for (int m = 16; m; m >>= 1) mx = fmaxf(mx, __shfl_xor(mx, m, 32));

  __shared__ float s_d2[WAVES_PER_BLOCK];
  __shared__ float s_mx[WAVES_PER_BLOCK];
  if (lane == 0) { s_d2[wave] = d2_t; s_mx[wave] = mx; }
  __syncthreads();
  if (threadIdx.x == 0) {
    float a = 0.0f, bmx = -INFINITY;
    for (int i = 0; i < WAVES_PER_BLOCK; ++i) {
      a += s_d2[i];
      bmx = fmaxf(bmx, s_mx[i]);
    }
    d2_partial[blockIdx.x] = a;
    atomicMax(gmax_key, flip_f32(bmx));  // exact op -> deterministic
  }
}

// ---- Kernel 2: per-row classification loss + physics penalty ----
__global__ __launch_bounds__(256) void k2_cls(
    const float* __restrict__ logits, const int* __restrict__ labels,
    const float* __restrict__ e_raw, const unsigned* __restrict__ gmax_key,
    float* __restrict__ nll_p, float* __restrict__ cnt_p,
    float* __restrict__ pen_p, int B) {
  const int row = blockIdx.x * blockDim.x + threadIdx.x;
  float nll = 0.0f, cnt = 0.0f, pen = 0.0f;
  if (row < B) {
    const float gmax = unflip_f32(*gmax_key);
    const v4f lg = ((const v4f*)logits)[row];
    // argmax, first-max tie-break (matches jnp.argmax)
    int cls = 0; float best = lg.x;
    if (lg.y > best) { best = lg.y; cls = 1; }
    if (lg.z > best) { best = lg.z; cls = 2; }
    if (lg.w > best) { best = lg.w; cls = 3; }
    // masked NLL via log-softmax
    const int lbl = labels[row];
    const bool valid = (lbl != -1);
    int sl = valid ? lbl : 0;
    sl = (sl < 0) ? 0 : ((sl > 3) ? 3 : sl);
    const float se = __expf(lg.x - best) + __expf(lg.y - best) +
                     __expf(lg.z - best) + __expf(lg.w - best);
    const float lse = best + __logf(se);
    const float lsel = (sl == 0) ? lg.x : (sl == 1) ? lg.y
                     : (sl == 2) ? lg.z : lg.w;
    if (valid) { nll = lse - lsel; cnt = 1.0f; }
    // physics penalty
    const float e = e_raw[row] / (20.0f * gmax);  // WIN * max(waveform)
    const bool isdef = (cls == 2) || (cls == 3);
    const float pd = (e < 0.1f) ? (0.1f - e) * (0.1f - e) : 0.0f;
    const float pc = (e > 0.5f) ? (e - 0.5f) * (e - 0.5f) : 0.0f;
    pen = isdef ? pd : pc;
  }
  __shared__ double sd[256 * 3];
  sd[threadIdx.x]       = (double)nll;
  sd[256 + threadIdx.x] = (double)cnt;
  sd[512 + threadIdx.x] = (double)pen;
  __syncthreads();
  for (int s = 128; s > 0; s >>= 1) {
    if (threadIdx.x < s) {
      sd[threadIdx.x]       += sd[threadIdx.x + s];
      sd[256 + threadIdx.x] += sd[256 + threadIdx.x + s];
      sd[512 + threadIdx.x] += sd[512 + threadIdx.x + s];
    }
    __syncthreads();
  }
  if (threadIdx.x == 0) {
    nll_p[blockIdx.x] = (float)sd[0];
    cnt_p[blockIdx.x] = (float)sd[256];
    pen_p[blockIdx.x] = (float)sd[512];
  }
}

// ---- Kernel 3: deterministic final reduction + loss assembly ----
__global__ __launch_bounds__(256) void k3_final(
    const float* __restrict__ d2_partial, int n1,
    const float* __restrict__ nll_p, const float* __restrict__ cnt_p,
    const float* __restrict__ pen_p, int n2,
    float* __restrict__ out, int B) {
  __shared__ double sd[256];
  const int t = threadIdx.x;
  double acc;

  acc = 0.0; for (int i = t; i < n1; i += 256) acc += (double)d2_partial[i];
  sd[t] = acc; __syncthreads();
  for (int s = 128; s > 0; s >>= 1) { if (t < s) sd[t] += sd[t + s]; __syncthreads(); }
  const double d2sum = sd[0]; __syncthreads();

  acc = 0.0; for (int i = t; i < n2; i += 256) acc += (double)nll_p[i];
  sd[t] = acc; __syncthreads();
  for (int s = 128; s > 0; s >>= 1) { if (t < s) sd[t] += sd[t + s]; __syncthreads(); }
  const double nll = sd[0]; __syncthreads();

  acc = 0.0; for (int i = t; i < n2; i += 256) acc += (double)cnt_p[i];
  sd[t] = acc; __syncthreads();
  for (int s = 128; s > 0; s >>= 1) { if (t < s) sd[t] += sd[t + s]; __syncthreads(); }
  const double cnt = sd[0]; __syncthreads();

  acc = 0.0; for (int i = t; i < n2; i += 256) acc += (double)pen_p[i];
  sd[t] = acc; __syncthreads();
  for (int s = 128; s > 0; s >>= 1) { if (t < s) sd[t] += sd[t + s]; __syncthreads(); }
  const double pen = sd[0];

  if (t == 0) {
    const double KF = 9900.0;  // 1/DT^2 - C^2/DX^2 = 10000 - 100
    const double wave_loss = (KF * KF) * d2sum / ((double)B * (double)LROW);
    const double cls_loss  = (cnt > 0.0) ? (nll / cnt) : 0.0;
    const double phys      = pen / (double)B;
    const double total     = cls_loss + 0.1 * wave_loss + 0.1 * phys;
    out[0] = (float)total;
    out[1] = (float)cls_loss;
    out[2] = (float)wave_loss;
    out[3] = (float)phys;
  }
}

extern "C" void kernel_launch(void* const* d_in, const int* in_sizes, int n_in,
                              void* d_out, int out_size, void* d_ws, size_t ws_size,
                              hipStream_t stream) {
  const float* wf     = (const float*)d_in[0];
  const float* logits = (const float*)d_in[1];
  const float* depth  = (const float*)d_in[2];
  const int*   labels = (const int*)d_in[3];
  const int B = in_sizes[3];

  // workspace carve-up (all regions fully rewritten every call)
  char* ws = (char*)d_ws;
  unsigned* gmax_key  = (unsigned*)ws;                         // 4 B (+pad)
  float*    e_raw     = (float*)(ws + 256);                    // B floats
  float*    d2_part   = (float*)(ws + 256 + (size_t)B * 4);    // K1_BLOCKS
  const int nblk2 = (B + 255) / 256;
  float* nll_p = d2_part + K1_BLOCKS;
  float* cnt_p = nll_p + nblk2;
  float* pen_p = cnt_p + nblk2;

  k0_init<<<1, 1, 0, stream>>>(gmax_key);
  k1_rows<<<K1_BLOCKS, K1_THREADS, 0, stream>>>(wf, depth, e_raw, d2_part,
                                                gmax_key, B);
  k2_cls<<<nblk2, 256, 0, stream>>>(logits, labels, e_raw, gmax_key,
                                    nll_p, cnt_p, pen_p, B);
  k3_final<<<1, 256, 0, stream>>>(d2_part, K1_BLOCKS, nll_p, cnt_p, pen_p,
                                  nblk2, (float*)d_out, B);
}